// FixedRadiusNNGraph_3487513444654
// MI455X (gfx1250) — compile-verified
//
#include <hip/hip_runtime.h>

typedef __attribute__((ext_vector_type(2))) float v2f;
typedef __attribute__((ext_vector_type(8))) float v8f;

#define RADIUS2 0.09f   // 0.3 * 0.3

constexpr int TILE = 16;
constexpr int WPB  = 4;   // waves per block (128 threads, wave32)

// One wave computes one 16x16 tile of dists/adj via a single
// V_WMMA_F32_16X16X4_F32.  K-slot 3 folds ||a||^2 into the MMA:
//   A[m][0..2] = a_m ,  A[m][3] = ||a_m||^2
//   B[0..2][n] = -2*b_n , B[3][n] = 1
//   => D[m][n] = ||a_m||^2 - 2 a_m.b_n ;  dists = D + ||b_n||^2 (lane-local).
//
// Templated on N so the 8 row offsets (r*N*4 bytes <= 229KB < 2^23) fold into
// the GLOBAL store 24-bit immediate offsets: only two 64-bit base addresses
// are computed per wave instead of sixteen.
template <int N>
__global__ __launch_bounds__(128) void frnn_dist_wmma(
    const float* __restrict__ pts,
    float* __restrict__ adj,
    float* __restrict__ dists)
{
    const int lane = threadIdx.x & 31;
    const int wave = threadIdx.x >> 5;
    const int lh   = lane & 15;   // M (for A) / N (for B) index within tile
    const int hi   = lane >> 4;   // 0: K=0,1 ; 1: K=2,3

    const int j0 = blockIdx.x * TILE;
    const int i0 = (blockIdx.y * WPB + wave) * TILE;
    const int b  = blockIdx.z;

    const float* __restrict__ pb = pts + (size_t)b * N * 3;

    // Row point (i0+lh) for the A tile (duplicated across half-waves).
    const float ax = pb[(i0 + lh) * 3 + 0];
    const float ay = pb[(i0 + lh) * 3 + 1];
    const float az = pb[(i0 + lh) * 3 + 2];
    const float sqa = ax * ax + ay * ay + az * az;

    // Column point (j0+lh) for the B tile.
    const float bx = pb[(j0 + lh) * 3 + 0];
    const float by = pb[(j0 + lh) * 3 + 1];
    const float bz = pb[(j0 + lh) * 3 + 2];
    const float sqb = bx * bx + by * by + bz * bz;

    // A-matrix 16x4 (MxK) VGPR layout: lanes 0-15 -> {K0,K1}, lanes 16-31 -> {K2,K3}.
    v2f A, Bm;
    A.x  = hi ? az  : ax;
    A.y  = hi ? sqa : ay;
    // B-matrix 4x16 (KxN), N striped over lanes, same K split per half-wave.
    Bm.x = -2.0f * (hi ? bz : bx);
    Bm.y = hi ? 1.0f : (-2.0f * by);

    v8f c = {0.f, 0.f, 0.f, 0.f, 0.f, 0.f, 0.f, 0.f};
    // 8 args: (neg_a, A, neg_b, B, c_mod, C, reuse_a, reuse_b)
    v8f d = __builtin_amdgcn_wmma_f32_16x16x4_f32(
        false, A, false, Bm, (short)0, c, false, false);

    // C/D layout: VGPR r -> row (r + 8*hi), col lh.  This lane's 8 elements
    // live on rows (i0 + 8*hi + r), all at column (j0 + lh): one base address
    // per plane, row offsets become store-immediate offsets.
    const size_t base =
        (size_t)b * N * N + (size_t)(i0 + 8 * hi) * N + (size_t)(j0 + lh);
    float* __restrict__ drow = dists + base;
    float* __restrict__ arow = adj + base;
#pragma unroll
    for (int r = 0; r < 8; ++r) {
        const float dist = d[r] + sqb;
        __builtin_nontemporal_store(dist, drow + (size_t)r * N);
        __builtin_nontemporal_store(dist <= RADIUS2 ? 1.0f : 0.0f,
                                    arow + (size_t)r * N);
    }
}

// Runtime-N fallback (identical math, dynamic strides).
__global__ __launch_bounds__(128) void frnn_dist_wmma_dyn(
    const float* __restrict__ pts,
    float* __restrict__ adj,
    float* __restrict__ dists,
    int N)
{
    const int lane = threadIdx.x & 31;
    const int wave = threadIdx.x >> 5;
    const int lh   = lane & 15;
    const int hi   = lane >> 4;

    const int j0 = blockIdx.x * TILE;
    const int i0 = (blockIdx.y * WPB + wave) * TILE;
    const int b  = blockIdx.z;

    const float* __restrict__ pb = pts + (size_t)b * N * 3;

    const float ax = pb[(i0 + lh) * 3 + 0];
    const float ay = pb[(i0 + lh) * 3 + 1];
    const float az = pb[(i0 + lh) * 3 + 2];
    const float sqa = ax * ax + ay * ay + az * az;

    const float bx = pb[(j0 + lh) * 3 + 0];
    const float by = pb[(j0 + lh) * 3 + 1];
    const float bz = pb[(j0 + lh) * 3 + 2];
    const float sqb = bx * bx + by * by + bz * bz;

    v2f A, Bm;
    A.x  = hi ? az  : ax;
    A.y  = hi ? sqa : ay;
    Bm.x = -2.0f * (hi ? bz : bx);
    Bm.y = hi ? 1.0f : (-2.0f * by);

    v8f c = {0.f, 0.f, 0.f, 0.f, 0.f, 0.f, 0.f, 0.f};
    v8f d = __builtin_amdgcn_wmma_f32_16x16x4_f32(
        false, A, false, Bm, (short)0, c, false, false);

    const size_t base =
        (size_t)b * N * N + (size_t)(i0 + 8 * hi) * N + (size_t)(j0 + lh);
    float* __restrict__ drow = dists + base;
    float* __restrict__ arow = adj + base;
#pragma unroll
    for (int r = 0; r < 8; ++r) {
        const float dist = d[r] + sqb;
        __builtin_nontemporal_store(dist, drow + (size_t)r * N);
        __builtin_nontemporal_store(dist <= RADIUS2 ? 1.0f : 0.0f,
                                    arow + (size_t)r * N);
    }
}

extern "C" void kernel_launch(void* const* d_in, const int* in_sizes, int n_in,
                              void* d_out, int out_size, void* d_ws, size_t ws_size,
                              hipStream_t stream) {
    const float* batch_points = (const float*)d_in[0];
    const float* batch_feats  = (const float*)d_in[1];
    // batch_len (d_in[2]) encodes uniform segments: B = len(batch_len) - 1.
    const int B = in_sizes[2] - 1;
    const int totalPts = in_sizes[0] / 3;
    const int N = totalPts / B;

    float* out = (float*)d_out;
    const size_t nn = (size_t)B * (size_t)N * (size_t)N;
    float* adj_out   = out;            // [B,N,N] bool -> 0.0/1.0
    float* dists_out = out + nn;       // [B,N,N]
    float* pts_out   = out + 2 * nn;   // [B,N,3] passthrough
    float* feats_out = pts_out + (size_t)B * N * 3;  // [B,N,F] passthrough

    dim3 block(32 * WPB);
    dim3 grid(N / TILE, N / (TILE * WPB), B);
    if (N == 8192) {
        hipLaunchKernelGGL(frnn_dist_wmma<8192>, grid, block, 0, stream,
                           batch_points, adj_out, dists_out);
    } else {
        hipLaunchKernelGGL(frnn_dist_wmma_dyn, grid, block, 0, stream,
                           batch_points, adj_out, dists_out, N);
    }

    hipMemcpyAsync(pts_out, batch_points, (size_t)in_sizes[0] * sizeof(float),
                   hipMemcpyDeviceToDevice, stream);
    hipMemcpyAsync(feats_out, batch_feats, (size_t)in_sizes[1] * sizeof(float),
                   hipMemcpyDeviceToDevice, stream);

    (void)n_in; (void)out_size; (void)d_ws; (void)ws_size;
}